// MultiHeadAttention_82162724372520
// MI455X (gfx1250) — compile-verified
//
#include <hip/hip_runtime.h>
#include <hip/hip_bf16.h>
#include <math.h>

// ---------------------------------------------------------------------------
// MHA forward for gfx1250 (MI455X), all matmuls via V_WMMA_F32_16X16X4_F32.
// Shapes: B=1, S=2048, D_MODEL=1024, H=16, DK=64.
// d_out = [ out (2048*1024 f32) | attn (16*2048*2048 f32) ]
// d_ws  = [ qf 8MiB | kf 8MiB | vT 8MiB | ctx 8MiB ]  (32 MiB)
// ---------------------------------------------------------------------------

typedef __attribute__((ext_vector_type(2))) float v2f;
typedef __attribute__((ext_vector_type(8))) float v8f;

#define S_LEN 2048
#define DM    1024
#define NH    16
#define DK    64

__device__ __forceinline__ v8f wmma4(v2f a, v2f b, v8f c) {
  // D = A(16x4,f32) x B(4x16,f32) + C(16x16,f32)
  return __builtin_amdgcn_wmma_f32_16x16x4_f32(false, a, false, b, (short)0, c,
                                               false, false);
}

// ---------------------------------------------------------------------------
// Generic projection GEMM: out[s,e] = sum_d X[s,d] * W[e,d] + b[e]
//   mode 0: write f32 [H][S][DK]          (q / k before RoPE)
//   mode 1: write f32 transposed vT [H][DK][S]
//   mode 2: write f32 [S][DM]             (final output projection)
// One wave computes a 16(M) x 64(N) strip; A-fragment reused across 4 N-tiles.
// Grid: 2048 waves = 256 blocks x 256 threads.
// ---------------------------------------------------------------------------
__global__ void proj_kernel(const float* __restrict__ X,
                            const float* __restrict__ W,
                            const float* __restrict__ bias,
                            float* __restrict__ out, int mode) {
  int tid  = blockIdx.x * blockDim.x + threadIdx.x;
  int w    = tid >> 5;
  int lane = tid & 31;
  int l16  = lane & 15;
  int hl   = lane >> 4;

  int mt = w >> 4;   // 0..127  M tile (16 rows of S)
  int ng = w & 15;   // 0..15   N group (64 cols of E)
  int s0 = mt * 16;
  int e0 = ng * 64;

  const float* Arow = X + (size_t)(s0 + l16) * DM + 2 * hl;
  const float* Brow[4];
#pragma unroll
  for (int n = 0; n < 4; ++n)
    Brow[n] = W + (size_t)(e0 + n * 16 + l16) * DM + 2 * hl;

  v8f acc[4] = {{0.f,0.f,0.f,0.f,0.f,0.f,0.f,0.f},
                {0.f,0.f,0.f,0.f,0.f,0.f,0.f,0.f},
                {0.f,0.f,0.f,0.f,0.f,0.f,0.f,0.f},
                {0.f,0.f,0.f,0.f,0.f,0.f,0.f,0.f}};

  for (int k = 0; k < DM; k += 4) {
    v2f a = *(const v2f*)(Arow + k);
#pragma unroll
    for (int n = 0; n < 4; ++n) {
      v2f b = *(const v2f*)(Brow[n] + k);
      acc[n] = wmma4(a, b, acc[n]);
    }
  }

#pragma unroll
  for (int n = 0; n < 4; ++n) {
    int   e  = e0 + n * 16 + l16;     // output column for this lane
    float bv = bias[e];
    if (mode == 1) {
      // vT[h][dk][s]: lane owns rows s0+8*hl..+7 contiguously -> 2x float4
      int hh = e >> 6, dk = e & 63;
      float* dst = out + ((size_t)hh * DK + dk) * S_LEN + s0 + 8 * hl;
      float4 lo = make_float4(acc[n][0]+bv, acc[n][1]+bv, acc[n][2]+bv, acc[n][3]+bv);
      float4 hi = make_float4(acc[n][4]+bv, acc[n][5]+bv, acc[n][6]+bv, acc[n][7]+bv);
      *(float4*)(dst)     = lo;
      *(float4*)(dst + 4) = hi;
    } else if (mode == 0) {
      int hh = e >> 6, dk = e & 63;
#pragma unroll
      for (int r = 0; r < 8; ++r) {
        int srow = s0 + r + 8 * hl;
        out[((size_t)hh * S_LEN + srow) * DK + dk] = acc[n][r] + bv;
      }
    } else {
#pragma unroll
      for (int r = 0; r < 8; ++r) {
        int srow = s0 + r + 8 * hl;
        out[(size_t)srow * DM + e] = acc[n][r] + bv;
      }
    }
  }
}

// ---------------------------------------------------------------------------
// RoPE, in place on qf and kf ([H][S][DK] layout).
// theta_i = 10000^(-2 i / D_MODEL)  (D_MODEL in denominator, per reference)
// idx bits: [0:4]=i (32 pairs), [5:15]=s, [16:19]=h, [20]=buffer select
// ---------------------------------------------------------------------------
__global__ void rope_kernel(float* __restrict__ qf, float* __restrict__ kf,
                            const int* __restrict__ pos) {
  int idx = blockIdx.x * blockDim.x + threadIdx.x;
  int i = idx & 31;
  int s = (idx >> 5) & (S_LEN - 1);
  int h = (idx >> 16) & (NH - 1);
  int which = idx >> 20;
  float* buf = which ? kf : qf;

  float p     = (float)pos[s];
  float theta = expf(-2.0f * (float)i * (9.2103403719761836f / (float)DM));
  float ang   = p * theta;
  float sn, cs;
  sincosf(ang, &sn, &cs);

  float* base = buf + ((size_t)h * S_LEN + s) * DK + 2 * i;
  float xe = base[0];
  float xo = base[1];
  base[0] = xe * cs - xo * sn;
  base[1] = xe * sn + xo * cs;
}

// ---------------------------------------------------------------------------
// Masked scaled scores: attn[h][q][k] = (q.k)/8 if k<=q else -1e9.
// One wave: (h, q-tile) x 4 consecutive k-tiles; A(q) fragment preloaded.
// Grid: 16*128*32 = 65536 waves = 8192 blocks x 256 threads.
// ---------------------------------------------------------------------------
__global__ void scores_kernel(const float* __restrict__ qf,
                              const float* __restrict__ kf,
                              float* __restrict__ attn) {
  int tid  = blockIdx.x * blockDim.x + threadIdx.x;
  int w    = tid >> 5;
  int lane = tid & 31;
  int l16  = lane & 15;
  int hl   = lane >> 4;

  int kg = w & 31;           // k-group of 4 tiles
  int qt = (w >> 5) & 127;   // q tile
  int h  = w >> 12;          // head
  int q0 = qt * 16;
  int k0b = kg * 64;

  // Preload full 16x64 A fragment for this wave (16 x v2f = 32 VGPRs)
  const float* Arow = qf + ((size_t)h * S_LEN + q0 + l16) * DK + 2 * hl;
  v2f afrag[16];
#pragma unroll
  for (int kk = 0; kk < 16; ++kk) afrag[kk] = *(const v2f*)(Arow + kk * 4);

  float* outb = attn + (size_t)h * S_LEN * S_LEN;

  for (int n = 0; n < 4; ++n) {
    int  k0       = k0b + n * 16;
    bool fullmask = (k0 > q0 + 15);
    v8f  acc = {0.f,0.f,0.f,0.f,0.f,0.f,0.f,0.f};
    if (!fullmask) {
      const float* Brow = kf + ((size_t)h * S_LEN + k0 + l16) * DK + 2 * hl;
#pragma unroll
      for (int kk = 0; kk < 16; ++kk) {
        v2f b = *(const v2f*)(Brow + kk * 4);
        acc = wmma4(afrag[kk], b, acc);
      }
    }
    int kpos = k0 + l16;
#pragma unroll
    for (int r = 0; r < 8; ++r) {
      int   qpos = q0 + r + 8 * hl;
      float v = (!fullmask && kpos <= qpos) ? acc[r] * 0.125f : -1.0e9f;
      outb[(size_t)qpos * S_LEN + kpos] = v;
    }
  }
}

// ---------------------------------------------------------------------------
// Row softmax over attn, in place. One 256-thread block per row (32768 rows).
// Masked entries (-1e9) underflow exp() to exact 0, matching the reference.
// ---------------------------------------------------------------------------
__global__ void softmax_kernel(float* __restrict__ attn) {
  float* p = attn + (size_t)blockIdx.x * S_LEN;
  int t = threadIdx.x;
  __shared__ float red[256];

  float vals[8];
  float mx = -INFINITY;
#pragma unroll
  for (int j = 0; j < 8; ++j) {
    vals[j] = p[t + j * 256];
    mx = fmaxf(mx, vals[j]);
  }
  red[t] = mx;
  __syncthreads();
  for (int s = 128; s > 0; s >>= 1) {
    if (t < s) red[t] = fmaxf(red[t], red[t + s]);
    __syncthreads();
  }
  mx = red[0];
  __syncthreads();

  float sum = 0.f;
#pragma unroll
  for (int j = 0; j < 8; ++j) {
    vals[j] = expf(vals[j] - mx);
    sum += vals[j];
  }
  red[t] = sum;
  __syncthreads();
  for (int s = 128; s > 0; s >>= 1) {
    if (t < s) red[t] += red[t + s];
    __syncthreads();
  }
  float inv = 1.0f / red[0];
#pragma unroll
  for (int j = 0; j < 8; ++j) p[t + j * 256] = vals[j] * inv;
}

// ---------------------------------------------------------------------------
// PV: ctx[q][h*64+d] = sum_k attn[h][q][k] * v[h][k][d], with V transposed in
// ws (vT[h][d][s]) so the B operand is K-contiguous. One wave per (h, q-tile)
// computes 16 x 64; K-loop stops at q0+16 (probs beyond causal edge are 0).
// Grid: 2048 waves = 256 blocks x 256 threads.
// ---------------------------------------------------------------------------
__global__ void pv_kernel(const float* __restrict__ attn,
                          const float* __restrict__ vT,
                          float* __restrict__ ctx) {
  int tid  = blockIdx.x * blockDim.x + threadIdx.x;
  int w    = tid >> 5;
  int lane = tid & 31;
  int l16  = lane & 15;
  int hl   = lane >> 4;

  int qt = w & 127;
  int h  = w >> 7;
  int q0 = qt * 16;

  const float* Arow = attn + ((size_t)h * S_LEN + q0 + l16) * S_LEN + 2 * hl;
  const float* Brow[4];
#pragma unroll
  for (int n = 0; n < 4; ++n)
    Brow[n] = vT + ((size_t)h * DK + n * 16 + l16) * S_LEN + 2 * hl;

  v8f acc[4] = {{0.f,0.f,0.f,0.f,0.f,0.f,0.f,0.f},
                {0.f,0.f,0.f,0.f,0.f,0.f,0.f,0.f},
                {0.f,0.f,0.f,0.f,0.f,0.f,0.f,0.f},
                {0.f,0.f,0.f,0.f,0.f,0.f,0.f,0.f}};

  int kmax = q0 + 16;  // multiple of 4; probs for k >= kmax are exactly 0
  for (int k = 0; k < kmax; k += 4) {
    v2f a = *(const v2f*)(Arow + k);
#pragma unroll
    for (int n = 0; n < 4; ++n) {
      v2f b = *(const v2f*)(Brow[n] + k);
      acc[n] = wmma4(a, b, acc[n]);
    }
  }

#pragma unroll
  for (int n = 0; n < 4; ++n) {
    int e = h * DK + n * 16 + l16;
#pragma unroll
    for (int r = 0; r < 8; ++r) {
      int srow = q0 + r + 8 * hl;
      ctx[(size_t)srow * DM + e] = acc[n][r];
    }
  }
}

// ---------------------------------------------------------------------------
extern "C" void kernel_launch(void* const* d_in, const int* in_sizes, int n_in,
                              void* d_out, int out_size, void* d_ws,
                              size_t ws_size, hipStream_t stream) {
  (void)in_sizes; (void)n_in; (void)out_size; (void)ws_size;

  const float* Q   = (const float*)d_in[0];
  const float* K   = (const float*)d_in[1];
  const float* V   = (const float*)d_in[2];
  const int*   pos = (const int*)  d_in[3];
  // d_in[4] = tril mask (bool) -> causal mask applied analytically
  const float* W_q = (const float*)d_in[5];
  const float* b_q = (const float*)d_in[6];
  const float* W_k = (const float*)d_in[7];
  const float* b_k = (const float*)d_in[8];
  const float* W_v = (const float*)d_in[9];
  const float* b_v = (const float*)d_in[10];
  const float* W_o = (const float*)d_in[11];
  const float* b_o = (const float*)d_in[12];

  float* out  = (float*)d_out;                                  // [2048][1024]
  float* attn = out + (size_t)S_LEN * DM;                       // [16][2048][2048]

  char*  ws  = (char*)d_ws;
  float* qf  = (float*)(ws);                                    // [16][2048][64]
  float* kf  = (float*)(ws + (size_t)8  * 1024 * 1024);         // [16][2048][64]
  float* vT  = (float*)(ws + (size_t)16 * 1024 * 1024);         // [16][64][2048]
  float* ctx = (float*)(ws + (size_t)24 * 1024 * 1024);         // [2048][1024]

  dim3 blk(256);

  // Projections (f32 WMMA GEMM): 2048 waves each
  proj_kernel<<<dim3(256), blk, 0, stream>>>(Q, W_q, b_q, qf, 0);
  proj_kernel<<<dim3(256), blk, 0, stream>>>(K, W_k, b_k, kf, 0);
  proj_kernel<<<dim3(256), blk, 0, stream>>>(V, W_v, b_v, vT, 1);

  // RoPE on q and k: 2*16*2048*32 = 2,097,152 threads
  rope_kernel<<<dim3(8192), blk, 0, stream>>>(qf, kf, pos);

  // Masked scaled scores into attn buffer: 65536 waves
  scores_kernel<<<dim3(8192), blk, 0, stream>>>(qf, kf, attn);

  // Row softmax in place: one block per row
  softmax_kernel<<<dim3(NH * S_LEN), blk, 0, stream>>>(attn);

  // attn @ V -> ctx: 2048 waves
  pv_kernel<<<dim3(256), blk, 0, stream>>>(attn, vT, ctx);

  // Output projection -> d_out
  proj_kernel<<<dim3(256), blk, 0, stream>>>(ctx, W_o, b_o, out, 2);
}